// HardNegativeSampler_50869592655084
// MI455X (gfx1250) — compile-verified
//
#include <hip/hip_runtime.h>
#include <hip/hip_bf16.h>

typedef float v2f __attribute__((ext_vector_type(2)));
typedef float v8f __attribute__((ext_vector_type(8)));

#define NUM_NEG  256
#define NUM_HARD 128
#define DIM      128

// ---------------------------------------------------------------------------
// Kernel A: cand_ids[p] = all_item_ids[cand_off[p]];
//           cand_emb[p,:] = l2norm(item_emb[cand_ids[p],:])
// One wave32 per candidate row; float4 per lane (32*4 = 128 elems).
// ---------------------------------------------------------------------------
__global__ __launch_bounds__(256) void prep_candidates(
    const float* __restrict__ item_emb, const int* __restrict__ all_item_ids,
    const int* __restrict__ cand_off, float* __restrict__ cand_emb,
    int* __restrict__ cand_ids, int P) {
  int w    = (blockIdx.x * blockDim.x + threadIdx.x) >> 5;
  int lane = threadIdx.x & 31;
  if (w >= P) return;
  int id = all_item_ids[cand_off[w]];
  const float4* src = (const float4*)(item_emb + (size_t)id * DIM);
  float4 v = src[lane];
  float s = v.x * v.x + v.y * v.y + v.z * v.z + v.w * v.w;
  #pragma unroll
  for (int off = 16; off > 0; off >>= 1) s += __shfl_xor(s, off, 32);
  float inv = 1.0f / fmaxf(sqrtf(s), 1e-8f);
  v.x *= inv; v.y *= inv; v.z *= inv; v.w *= inv;
  ((float4*)(cand_emb + (size_t)w * DIM))[lane] = v;
  if (lane == 0) cand_ids[w] = id;
}

// ---------------------------------------------------------------------------
// Kernel B: sim[T,P] = Q @ cand_emb^T with positive masking fused.
// 256 threads = 8 waves; block tile 32(M) x 64(N); wave tile 16x16.
// Full-precision f32 WMMA (16x16x4), K = 128 -> 32 WMMA ops per wave tile.
// ---------------------------------------------------------------------------
__global__ __launch_bounds__(256) void sim_wmma(
    const float* __restrict__ q, const float* __restrict__ cand_emb,
    const int* __restrict__ pos_ids, const int* __restrict__ cand_ids,
    float* __restrict__ sim, int T, int P) {
  __shared__ float As[32 * DIM];   // 16 KB
  __shared__ float Bs[64 * DIM];   // 32 KB
  const int tid = threadIdx.x;
  const int rowBase = blockIdx.y * 32;
  const int colBase = blockIdx.x * 64;

  // Stage tiles to LDS (float4 vectorized, fully coalesced).
  const float4* q4 = (const float4*)q;
  const float4* b4 = (const float4*)cand_emb;
  float4* As4 = (float4*)As;
  float4* Bs4 = (float4*)Bs;
  #pragma unroll
  for (int i = tid; i < 32 * (DIM / 4); i += 256)
    As4[i] = q4[(size_t)(rowBase + (i >> 5)) * (DIM / 4) + (i & 31)];
  #pragma unroll
  for (int i = tid; i < 64 * (DIM / 4); i += 256)
    Bs4[i] = b4[(size_t)(colBase + (i >> 5)) * (DIM / 4) + (i & 31)];
  __syncthreads();

  const int lane = tid & 31;
  const int wv   = tid >> 5;
  const int m0   = (wv & 1) * 16;   // wave tile origin in M (0 or 16)
  const int n0   = (wv >> 1) * 16;  // wave tile origin in N (0..48)
  const int half = lane >> 4;       // 0: lanes 0-15, 1: lanes 16-31
  const int lm   = lane & 15;

  v8f c = {};
#if defined(__gfx1250__) && __has_builtin(__builtin_amdgcn_wmma_f32_16x16x4_f32)
  // ISA 7.12.2 f32 layouts:
  //   A 16x4:  VGPR0 = (M=lane%16, K = half*2), VGPR1 = K = half*2+1
  //   B 4x16:  VGPR0 = (K = half*2, N=lane%16), VGPR1 = K = half*2+1
  const int kA = half * 2;
  #pragma unroll
  for (int k0 = 0; k0 < DIM; k0 += 4) {
    v2f a, b;
    a.x = As[(m0 + lm) * DIM + k0 + kA];
    a.y = As[(m0 + lm) * DIM + k0 + kA + 1];
    b.x = Bs[(n0 + lm) * DIM + k0 + kA];
    b.y = Bs[(n0 + lm) * DIM + k0 + kA + 1];
    c = __builtin_amdgcn_wmma_f32_16x16x4_f32(
        /*neg_a=*/false, a, /*neg_b=*/false, b,
        /*c_mod=*/(short)0, c, /*reuse_a=*/false, /*reuse_b=*/false);
  }
#else
  // VALU fallback with identical C layout (M = r + half*8, N = lane%16).
  #pragma unroll
  for (int r = 0; r < 8; ++r) {
    const int m = m0 + r + half * 8;
    const int n = n0 + lm;
    float acc = 0.0f;
    for (int k = 0; k < DIM; ++k) acc = fmaf(As[m * DIM + k], Bs[n * DIM + k], acc);
    c[r] = acc;
  }
#endif

  // Epilogue: C/D 16x16 f32 layout -> VGPR r holds (M = r + half*8, N = lane%16).
  const int n   = colBase + n0 + lm;
  const int cid = cand_ids[n];
  #pragma unroll
  for (int r = 0; r < 8; ++r) {
    const int m = rowBase + m0 + r + half * 8;
    float v = c[r];
    if (pos_ids[m] == cid) v = -__builtin_inff();
    sim[(size_t)m * P + n] = v;
  }
}

// ---------------------------------------------------------------------------
// Kernel C: per-row top-128 via in-LDS bitonic sort of packed 64-bit keys
//   key = monotone(f32 value) << 32 | ~index   (ties -> lowest index first,
//   matching jax.lax.top_k). Emits hard ids + copies hard embeddings.
// ---------------------------------------------------------------------------
__global__ __launch_bounds__(256) void topk_hard(
    const float* __restrict__ sim, const int* __restrict__ cand_ids,
    const float* __restrict__ cand_emb, float* __restrict__ out_ids,
    float* __restrict__ out_emb, int P) {
  const int N = 4096;  // P
  __shared__ unsigned long long keys[4096];  // 32 KB of 320 KB WGP LDS
  const int r   = blockIdx.x;
  const int tid = threadIdx.x;

  for (int i = tid; i < N; i += 256) {
    unsigned u = __float_as_uint(sim[(size_t)r * P + i]);
    u = (u & 0x80000000u) ? ~u : (u | 0x80000000u);  // order-preserving map
    keys[i] = ((unsigned long long)u << 32) | (unsigned)(~(unsigned)i);
  }
  __syncthreads();

  // Bitonic sort, descending.
  for (int k = 2; k <= N; k <<= 1) {
    for (int j = k >> 1; j > 0; j >>= 1) {
      for (int i = tid; i < N; i += 256) {
        const int ixj = i ^ j;
        if (ixj > i) {
          unsigned long long a = keys[i], b = keys[ixj];
          const bool up = ((i & k) == 0);
          if (up ? (a < b) : (a > b)) { keys[i] = b; keys[ixj] = a; }
        }
      }
      __syncthreads();
    }
  }

  for (int j = tid; j < NUM_HARD; j += 256) {
    const unsigned idx = ~(unsigned)(keys[j] & 0xFFFFFFFFull);
    out_ids[(size_t)r * NUM_NEG + j] = (float)cand_ids[idx];
  }
  for (int t = tid; t < NUM_HARD * DIM; t += 256) {
    const int j = t >> 7, dcol = t & (DIM - 1);
    const unsigned idx = ~(unsigned)(keys[j] & 0xFFFFFFFFull);
    out_emb[((size_t)r * NUM_NEG + j) * DIM + dcol] =
        cand_emb[(size_t)idx * DIM + dcol];
  }
}

// ---------------------------------------------------------------------------
// Kernel D: random negatives — one wave32 per (row, j): gather + l2norm + store.
// This kernel is the bandwidth floor (256 MB random read + 256 MB write).
// ---------------------------------------------------------------------------
__global__ __launch_bounds__(256) void random_negs(
    const float* __restrict__ item_emb, const int* __restrict__ all_item_ids,
    const int* __restrict__ rand_off, float* __restrict__ out_ids,
    float* __restrict__ out_emb, int T, int NR) {
  int w    = (blockIdx.x * blockDim.x + threadIdx.x) >> 5;
  int lane = threadIdx.x & 31;
  if (w >= T * NR) return;
  const int r = w / NR, j = w - r * NR;
  const int id = all_item_ids[rand_off[(size_t)r * NR + j]];
  const float4* src = (const float4*)(item_emb + (size_t)id * DIM);
  float4 v = src[lane];
  float s = v.x * v.x + v.y * v.y + v.z * v.z + v.w * v.w;
  #pragma unroll
  for (int off = 16; off > 0; off >>= 1) s += __shfl_xor(s, off, 32);
  const float inv = 1.0f / fmaxf(sqrtf(s), 1e-8f);
  v.x *= inv; v.y *= inv; v.z *= inv; v.w *= inv;
  ((float4*)(out_emb + ((size_t)r * NUM_NEG + NUM_HARD + j) * DIM))[lane] = v;
  if (lane == 0) out_ids[(size_t)r * NUM_NEG + NUM_HARD + j] = (float)id;
}

// ---------------------------------------------------------------------------
extern "C" void kernel_launch(void* const* d_in, const int* in_sizes, int n_in,
                              void* d_out, int out_size, void* d_ws, size_t ws_size,
                              hipStream_t stream) {
  const int*   pos_ids      = (const int*)d_in[0];
  const float* q            = (const float*)d_in[1];
  const float* item_emb     = (const float*)d_in[2];
  const int*   all_item_ids = (const int*)d_in[3];
  const int*   cand_off     = (const int*)d_in[4];
  const int*   rand_off     = (const int*)d_in[5];
  const int T  = in_sizes[0];        // 4096
  const int P  = in_sizes[4];        // 4096
  const int NR = in_sizes[5] / T;    // 128

  // Workspace: cand_emb[P*128] | sim[T*P] | cand_ids[P]
  char* ws = (char*)d_ws;
  float* cand_emb = (float*)ws;
  float* sim      = (float*)(ws + (size_t)P * DIM * sizeof(float));
  int*   cand_ids = (int*)(ws + (size_t)P * DIM * sizeof(float) +
                                (size_t)T * P * sizeof(float));

  float* out_ids = (float*)d_out;                 // [T, 256] ids as f32
  float* out_emb = out_ids + (size_t)T * NUM_NEG; // [T, 256, 128]

  prep_candidates<<<(P + 7) / 8, 256, 0, stream>>>(item_emb, all_item_ids,
                                                   cand_off, cand_emb, cand_ids, P);
  dim3 g(P / 64, T / 32);
  sim_wmma<<<g, 256, 0, stream>>>(q, cand_emb, pos_ids, cand_ids, sim, T, P);
  topk_hard<<<T, 256, 0, stream>>>(sim, cand_ids, cand_emb, out_ids, out_emb, P);
  random_negs<<<(T * NR + 7) / 8, 256, 0, stream>>>(item_emb, all_item_ids,
                                                    rand_off, out_ids, out_emb, T, NR);
}